// SubgraphGNNKernel_36661840838787
// MI455X (gfx1250) — compile-verified
//
#include <hip/hip_runtime.h>
#include <math.h>

// ---- problem constants (match reference) ----
#define NBASE 20000
#define NC    400000
#define ECNT  2000000
#define DD    80        // conv width
#define NOUT  64

typedef float v2f __attribute__((ext_vector_type(2)));
typedef float v8f __attribute__((ext_vector_type(8)));

// V_WMMA_F32_16X16X4_F32 : D(16x16 f32) = A(16x4 f32) x B(4x16 f32) + C
__device__ __forceinline__ v8f wmma4(v2f a, v2f b, v8f c) {
  return __builtin_amdgcn_wmma_f32_16x16x4_f32(
      /*neg_a=*/false, a, /*neg_b=*/false, b,
      /*c_mod=*/(short)0, c, /*reuse_a=*/false, /*reuse_b=*/false);
}

__device__ __forceinline__ void atomAdd(float* p, float v) {
  __hip_atomic_fetch_add(p, v, __ATOMIC_RELAXED, __HIP_MEMORY_SCOPE_AGENT);
}

// ---------------- zero fill (float4) ----------------
__global__ void k_zero4(float* __restrict__ p, int n4) {
  int t = blockIdx.x * blockDim.x + threadIdx.x;
  if (t < n4) ((float4*)p)[t] = make_float4(0.f, 0.f, 0.f, 0.f);
}

// ---------------- build h = concat(x[nm], hop_table[hi+1]) ----------------
// one float4 per thread: 20 quads per node (16 from x-row, 4 from hop row)
__global__ void k_build_h(const float* __restrict__ x,
                          const float* __restrict__ hop_table,
                          const int* __restrict__ nm,
                          const int* __restrict__ hi,
                          float* __restrict__ h) {
  int t = blockIdx.x * blockDim.x + threadIdx.x;
  if (t >= NC * 20) return;
  int node = t / 20, q = t - node * 20;
  float4 v;
  if (q < 16) {
    v = ((const float4*)(x + (long)nm[node] * 64))[q];
  } else {
    v = ((const float4*)(hop_table + (long)(hi[node] + 1) * 16))[q - 16];
  }
  ((float4*)(h + (long)node * DD))[q] = v;
}

// ---------------- GINE edge phase: agg[dst] += relu(h[src] + ea[em]) ------
// one float4 (a twentieth of an 80-float row) per thread: 20 quads per edge
__global__ void k_edge(const float* __restrict__ h,
                       const float* __restrict__ edge_attr,
                       const int* __restrict__ ei_src,
                       const int* __restrict__ ei_dst,
                       const int* __restrict__ emap,
                       float* __restrict__ agg) {
  int t = blockIdx.x * blockDim.x + threadIdx.x;
  if (t >= ECNT * 20) return;
  int e = t / 20, q = t - e * 20;
  int s = ei_src[e];
  int d = ei_dst[e];
  int m = emap[e];
  float4 hv = ((const float4*)(h + (long)s * DD))[q];
  float4 ev = ((const float4*)(edge_attr + (long)m * DD))[q];
  float mx = fmaxf(hv.x + ev.x, 0.f);
  float my = fmaxf(hv.y + ev.y, 0.f);
  float mz = fmaxf(hv.z + ev.z, 0.f);
  float mw = fmaxf(hv.w + ev.w, 0.f);
  float* base = agg + (long)d * DD + q * 4;
  atomAdd(base + 0, mx);
  atomAdd(base + 1, my);
  atomAdd(base + 2, mz);
  atomAdd(base + 3, mw);
}

// ---------------- GINE node phase: h = relu(g*((h+agg)@W)+b) + h ----------
// wave per 16-row tile; 5 N-tiles of 16, K=80 in 20 chunks of 4 (f32 WMMA)
// NOTE: h/hout are the SAME buffer (in-place) -> no __restrict__ on them, so
// the compiler must keep the residual load before the store at each address.
__global__ void k_gine_node(const float* h,
                            const float* __restrict__ agg,
                            const float* __restrict__ W,   // [80,80] row-major
                            const float* __restrict__ g,
                            const float* __restrict__ b,
                            float* hout) {
  int wave = (blockIdx.x * blockDim.x + threadIdx.x) >> 5;
  int lane = threadIdx.x & 31;
  if (wave >= NC / 16) return;
  long r0 = (long)wave * 16;
  int m = lane & 15;
  int kh = (lane >> 4) * 2;   // 0 or 2
  v8f acc[5];
#pragma unroll
  for (int t = 0; t < 5; ++t) acc[t] = (v8f)0.f;

#pragma unroll 4
  for (int kc = 0; kc < 20; ++kc) {
    int k0 = kc * 4 + kh;
    long ai = (r0 + m) * DD + k0;
    float2 hv = *(const float2*)(h + ai);
    float2 av = *(const float2*)(agg + ai);
    v2f a;
    a.x = hv.x + av.x;
    a.y = hv.y + av.y;
#pragma unroll
    for (int t = 0; t < 5; ++t) {
      int n = 16 * t + m;
      v2f bb;
      bb.x = W[(long)k0 * DD + n];
      bb.y = W[(long)(k0 + 1) * DD + n];
      acc[t] = wmma4(a, bb, acc[t]);
    }
  }
  // epilogue: eval-BN + ReLU + residual, in place
  int hoff = (lane >> 4) * 8;
#pragma unroll
  for (int v = 0; v < 8; ++v) {
    long r = r0 + v + hoff;
#pragma unroll
    for (int t = 0; t < 5; ++t) {
      int c = 16 * t + m;
      float val = g[c] * acc[t][v] + b[c];
      val = fmaxf(val, 0.f);
      float res = h[r * DD + c];        // read residual BEFORE in-place store
      hout[r * DD + c] = val + res;
    }
  }
}

// ---------------- comb = h @ oe_w + oe_b  ([NC,80]x[80,64]) ----------------
__global__ void k_comb(const float* __restrict__ h,
                       const float* __restrict__ W,  // [80,64]
                       const float* __restrict__ bias,
                       float* __restrict__ comb) {
  int wave = (blockIdx.x * blockDim.x + threadIdx.x) >> 5;
  int lane = threadIdx.x & 31;
  if (wave >= NC / 16) return;
  long r0 = (long)wave * 16;
  int m = lane & 15;
  int kh = (lane >> 4) * 2;
  v8f acc[4];
#pragma unroll
  for (int t = 0; t < 4; ++t) acc[t] = (v8f)0.f;

#pragma unroll 4
  for (int kc = 0; kc < 20; ++kc) {
    int k0 = kc * 4 + kh;
    float2 hv = *(const float2*)(h + (r0 + m) * DD + k0);
    v2f a; a.x = hv.x; a.y = hv.y;
#pragma unroll
    for (int t = 0; t < 4; ++t) {
      int n = 16 * t + m;
      v2f bb;
      bb.x = W[(long)k0 * NOUT + n];
      bb.y = W[(long)(k0 + 1) * NOUT + n];
      acc[t] = wmma4(a, bb, acc[t]);
    }
  }
  int hoff = (lane >> 4) * 8;
#pragma unroll
  for (int v = 0; v < 8; ++v) {
    long r = r0 + v + hoff;
#pragma unroll
    for (int t = 0; t < 4; ++t) {
      int c = 16 * t + m;
      comb[r * NOUT + c] = acc[t][v] + bias[c];
    }
  }
}

// ------- precompute 20-row gate tables: sigmoid(hop_table[k] @ W + b) ------
__global__ void k_gates(const float* __restrict__ hop_table,
                        const float* __restrict__ gcw, const float* __restrict__ gcb,
                        const float* __restrict__ gsw, const float* __restrict__ gsb,
                        const float* __restrict__ gxw, const float* __restrict__ gxb,
                        float* __restrict__ gateC, float* __restrict__ gateS,
                        float* __restrict__ gateX) {
  int t = blockIdx.x * blockDim.x + threadIdx.x;
  if (t >= 3 * 20 * NOUT) return;
  int which = t / (20 * NOUT);
  int rem = t - which * 20 * NOUT;
  int k = rem / NOUT, n = rem - k * NOUT;
  const float* W = (which == 0) ? gcw : (which == 1) ? gsw : gxw;
  const float* B = (which == 0) ? gcb : (which == 1) ? gsb : gxb;
  float s = B[n];
  const float* hr = hop_table + k * 16;
#pragma unroll
  for (int j = 0; j < 16; ++j) s += hr[j] * W[j * NOUT + n];
  float out = 1.f / (1.f + expf(-s));
  float* dst = (which == 0) ? gateC : (which == 1) ? gateS : gateX;
  dst[k * NOUT + n] = out;
}

// ---------------- segment counts ----------------
__global__ void k_counts(const int* __restrict__ batch, const int* __restrict__ nm,
                         float* __restrict__ cnt_b, float* __restrict__ cnt_m) {
  int i = blockIdx.x * blockDim.x + threadIdx.x;
  if (i >= NC) return;
  atomAdd(&cnt_b[batch[i]], 1.f);
  atomAdd(&cnt_m[nm[i]], 1.f);
}

// ---- pooling: two fused 64x64 GEMMs + gates + scatter into 3 pools --------
__global__ void k_pool(const float* __restrict__ comb,
                       const float* __restrict__ sub_w, const float* __restrict__ sub_b,
                       const float* __restrict__ sub_g, const float* __restrict__ sub_bt,
                       const float* __restrict__ ctx_w, const float* __restrict__ ctx_b,
                       const float* __restrict__ ctx_g, const float* __restrict__ ctx_bt,
                       const float* __restrict__ gateC, const float* __restrict__ gateS,
                       const float* __restrict__ gateX,
                       const int* __restrict__ batch, const int* __restrict__ nm,
                       const int* __restrict__ hi,
                       float* __restrict__ centroid, float* __restrict__ sub_pool,
                       float* __restrict__ ctx_pool) {
  int wave = (blockIdx.x * blockDim.x + threadIdx.x) >> 5;
  int lane = threadIdx.x & 31;
  if (wave >= NC / 16) return;
  long r0 = (long)wave * 16;
  int m = lane & 15;
  int kh = (lane >> 4) * 2;
  v8f as[4], ac[4];
#pragma unroll
  for (int t = 0; t < 4; ++t) { as[t] = (v8f)0.f; ac[t] = (v8f)0.f; }

#pragma unroll 4
  for (int kc = 0; kc < 16; ++kc) {
    int k0 = kc * 4 + kh;
    float2 cv = *(const float2*)(comb + (r0 + m) * NOUT + k0);
    v2f a; a.x = cv.x; a.y = cv.y;
#pragma unroll
    for (int t = 0; t < 4; ++t) {
      int n = 16 * t + m;
      v2f bs, bc;
      bs.x = sub_w[(long)k0 * NOUT + n];
      bs.y = sub_w[(long)(k0 + 1) * NOUT + n];
      bc.x = ctx_w[(long)k0 * NOUT + n];
      bc.y = ctx_w[(long)(k0 + 1) * NOUT + n];
      as[t] = wmma4(a, bs, as[t]);
      ac[t] = wmma4(a, bc, ac[t]);
    }
  }
  int hoff = (lane >> 4) * 8;
#pragma unroll
  for (int v = 0; v < 8; ++v) {
    long r = r0 + v + hoff;
    int bi = batch[r];
    int mi = nm[r];
    int hh = hi[r] + 1;
    bool cent = (bi == mi);
#pragma unroll
    for (int t = 0; t < 4; ++t) {
      int c = 16 * t + m;
      float sv = fmaxf(sub_g[c] * (as[t][v] + sub_b[c]) + sub_bt[c], 0.f) * gateS[hh * NOUT + c];
      atomAdd(&sub_pool[(long)bi * NOUT + c], sv);
      float cvv = fmaxf(ctx_g[c] * (ac[t][v] + ctx_b[c]) + ctx_bt[c], 0.f) * gateX[hh * NOUT + c];
      atomAdd(&ctx_pool[(long)mi * NOUT + c], cvv);
      if (cent) {
        atomAdd(&centroid[(long)bi * NOUT + c], comb[r * NOUT + c] * gateC[hh * NOUT + c]);
      }
    }
  }
}

// ---- final: out = fo_g*((centroid+sub/cb+ctx/cm) @ fo_w + fo_b) + fo_bt ---
__global__ void k_final(const float* __restrict__ centroid,
                        const float* __restrict__ sub_pool,
                        const float* __restrict__ ctx_pool,
                        const float* __restrict__ cnt_b,
                        const float* __restrict__ cnt_m,
                        const float* __restrict__ W,  // fo_w [64,64]
                        const float* __restrict__ fb,
                        const float* __restrict__ fg,
                        const float* __restrict__ fbt,
                        float* __restrict__ out) {
  int wave = (blockIdx.x * blockDim.x + threadIdx.x) >> 5;
  int lane = threadIdx.x & 31;
  if (wave >= NBASE / 16) return;
  long r0 = (long)wave * 16;
  int m = lane & 15;
  int kh = (lane >> 4) * 2;
  long row = r0 + m;
  float icb = 1.f / fmaxf(cnt_b[row], 1.f);
  float icm = 1.f / fmaxf(cnt_m[row], 1.f);
  v8f acc[4];
#pragma unroll
  for (int t = 0; t < 4; ++t) acc[t] = (v8f)0.f;

#pragma unroll 4
  for (int kc = 0; kc < 16; ++kc) {
    int k0 = kc * 4 + kh;
    long i0 = row * NOUT + k0;
    v2f a;
    a.x = centroid[i0] + sub_pool[i0] * icb + ctx_pool[i0] * icm;
    a.y = centroid[i0 + 1] + sub_pool[i0 + 1] * icb + ctx_pool[i0 + 1] * icm;
#pragma unroll
    for (int t = 0; t < 4; ++t) {
      int n = 16 * t + m;
      v2f bb;
      bb.x = W[(long)k0 * NOUT + n];
      bb.y = W[(long)(k0 + 1) * NOUT + n];
      acc[t] = wmma4(a, bb, acc[t]);
    }
  }
  int hoff = (lane >> 4) * 8;
#pragma unroll
  for (int v = 0; v < 8; ++v) {
    long r = r0 + v + hoff;
#pragma unroll
    for (int t = 0; t < 4; ++t) {
      int c = 16 * t + m;
      out[r * NOUT + c] = fg[c] * (acc[t][v] + fb[c]) + fbt[c];
    }
  }
}

// =========================================================================
extern "C" void kernel_launch(void* const* d_in, const int* in_sizes, int n_in,
                              void* d_out, int out_size, void* d_ws, size_t ws_size,
                              hipStream_t stream) {
  const float* x        = (const float*)d_in[0];
  const float* edge_att = (const float*)d_in[1];
  const float* hop_tab  = (const float*)d_in[2];
  const float* conv_w1  = (const float*)d_in[3];
  const float* bn1_g    = (const float*)d_in[4];
  const float* bn1_b    = (const float*)d_in[5];
  const float* conv_w2  = (const float*)d_in[6];
  const float* bn2_g    = (const float*)d_in[7];
  const float* bn2_b    = (const float*)d_in[8];
  const float* oe_w     = (const float*)d_in[9];
  const float* oe_b     = (const float*)d_in[10];
  const float* sub_w    = (const float*)d_in[11];
  const float* sub_b    = (const float*)d_in[12];
  const float* sub_g    = (const float*)d_in[13];
  const float* sub_bt   = (const float*)d_in[14];
  const float* ctx_w    = (const float*)d_in[15];
  const float* ctx_b    = (const float*)d_in[16];
  const float* ctx_g    = (const float*)d_in[17];
  const float* ctx_bt   = (const float*)d_in[18];
  const float* gc_w     = (const float*)d_in[19];
  const float* gc_b     = (const float*)d_in[20];
  const float* gs_w     = (const float*)d_in[21];
  const float* gs_b     = (const float*)d_in[22];
  const float* gx_w     = (const float*)d_in[23];
  const float* gx_b     = (const float*)d_in[24];
  const float* fo_w     = (const float*)d_in[25];
  const float* fo_b     = (const float*)d_in[26];
  const float* fo_g     = (const float*)d_in[27];
  const float* fo_bt    = (const float*)d_in[28];
  const int* nm         = (const int*)d_in[29];
  const int* emap       = (const int*)d_in[30];
  const int* eidx       = (const int*)d_in[31];  // [2, EC]
  const int* batch      = (const int*)d_in[32];
  const int* hind       = (const int*)d_in[33];

  float* out = (float*)d_out;

  // ---- workspace layout (floats) ----
  float* ws = (float*)d_ws;
  float* h        = ws;                         // NC*80 = 32,000,000
  float* agg      = h + (long)NC * DD;          // NC*80 = 32,000,000
  float* comb     = agg;                        // alias: comb needed only after agg retired
  float* base2    = agg + (long)NC * DD;
  float* centroid = base2;                      // N*64
  float* sub_pool = centroid + (long)NBASE * NOUT;
  float* ctx_pool = sub_pool + (long)NBASE * NOUT;
  float* cnt_b    = ctx_pool + (long)NBASE * NOUT;
  float* cnt_m    = cnt_b + NBASE;
  float* gateC    = cnt_m + NBASE;              // 20*64 each
  float* gateS    = gateC + 20 * NOUT;
  float* gateX    = gateS + 20 * NOUT;

  const int TPB = 256;
  const int* ei_src = eidx;
  const int* ei_dst = eidx + ECNT;

  // 1. build h
  k_build_h<<<(NC * 20 + TPB - 1) / TPB, TPB, 0, stream>>>(x, hop_tab, nm, hind, h);

  // ---- layer 1 ----
  k_zero4<<<(NC * DD / 4 + TPB - 1) / TPB, TPB, 0, stream>>>(agg, NC * DD / 4);
  k_edge<<<(ECNT * 20 + TPB - 1) / TPB, TPB, 0, stream>>>(h, edge_att, ei_src, ei_dst, emap, agg);
  k_gine_node<<<(NC / 16 * 32 + TPB - 1) / TPB, TPB, 0, stream>>>(h, agg, conv_w1, bn1_g, bn1_b, h);

  // ---- layer 2 ----
  k_zero4<<<(NC * DD / 4 + TPB - 1) / TPB, TPB, 0, stream>>>(agg, NC * DD / 4);
  k_edge<<<(ECNT * 20 + TPB - 1) / TPB, TPB, 0, stream>>>(h, edge_att, ei_src, ei_dst, emap, agg);
  k_gine_node<<<(NC / 16 * 32 + TPB - 1) / TPB, TPB, 0, stream>>>(h, agg, conv_w2, bn2_g, bn2_b, h);

  // ---- output encoder (comb aliases agg; agg no longer needed) ----
  k_comb<<<(NC / 16 * 32 + TPB - 1) / TPB, TPB, 0, stream>>>(h, oe_w, oe_b, comb);

  // ---- gates (only 20 distinct hop rows) ----
  k_gates<<<(3 * 20 * NOUT + TPB - 1) / TPB, TPB, 0, stream>>>(
      hop_tab, gc_w, gc_b, gs_w, gs_b, gx_w, gx_b, gateC, gateS, gateX);

  // ---- zero pools + counts (contiguous region) ----
  {
    int nfl = 3 * NBASE * NOUT + 2 * NBASE;  // 3,880,000 (divisible by 4)
    k_zero4<<<(nfl / 4 + TPB - 1) / TPB, TPB, 0, stream>>>(centroid, nfl / 4);
  }

  // ---- counts + pooling ----
  k_counts<<<(NC + TPB - 1) / TPB, TPB, 0, stream>>>(batch, nm, cnt_b, cnt_m);
  k_pool<<<(NC / 16 * 32 + TPB - 1) / TPB, TPB, 0, stream>>>(
      comb, sub_w, sub_b, sub_g, sub_bt, ctx_w, ctx_b, ctx_g, ctx_bt,
      gateC, gateS, gateX, batch, nm, hind, centroid, sub_pool, ctx_pool);

  // ---- final encoder ----
  k_final<<<(NBASE / 16 * 32 + TPB - 1) / TPB, TPB, 0, stream>>>(
      centroid, sub_pool, ctx_pool, cnt_b, cnt_m, fo_w, fo_b, fo_g, fo_bt, out);
}